// DeltaNet_31877247271548
// MI455X (gfx1250) — compile-verified
//
#include <hip/hip_runtime.h>
#include <hip/hip_bf16.h>

typedef __attribute__((ext_vector_type(2))) float v2f;
typedef __attribute__((ext_vector_type(8))) float v8f;

#define CHUNKC 32
#define DK 256
#define DVT 64            // dv slice per workgroup
#define LSEQ 4096
#define NCHUNK (LSEQ / CHUNKC)
#define QLD 260           // padded row stride for 32x256 arrays (4 mod 64 banks)
#define VLD 68            // padded row stride for 32x64 arrays
#define ALD 36            // padded row stride for 32x32 arrays

__device__ __forceinline__ v8f wmma4(float ax, float ay, float bx, float by, v8f acc) {
  v2f a; a.x = ax; a.y = ay;
  v2f b; b.x = bx; b.y = by;
  return __builtin_amdgcn_wmma_f32_16x16x4_f32(false, a, false, b, (short)0, acc, false, false);
}

__device__ __forceinline__ float wsum32(float x) {
#pragma unroll
  for (int m = 16; m >= 1; m >>= 1) x += __shfl_xor(x, m, 32);
  return x;
}

__global__ void __launch_bounds__(256)
deltanet_chunk_kernel(const float* __restrict__ qg, const float* __restrict__ kg,
                      const float* __restrict__ vg, const float* __restrict__ bg,
                      float* __restrict__ outg, float* __restrict__ Sg)
{
  extern __shared__ float lds[];
  float* S     = lds;                      // 256*64 (row stride 64; row-contiguous access only)
  float* qs    = S    + DK * DVT;          // 32 x QLD
  float* ks    = qs   + CHUNKC * QLD;      // 32 x QLD
  float* ws    = ks   + CHUNKC * QLD;      // 32 x QLD
  float* vs    = ws   + CHUNKC * QLD;      // 32 x VLD
  float* us    = vs   + CHUNKC * VLD;      // 32 x VLD
  float* un    = us   + CHUNKC * VLD;      // 32 x VLD
  float* Al    = un   + CHUNKC * VLD;      // 32 x ALD  (A; upper-right quad reused for P; then attn)
  float* Tl    = Al   + CHUNKC * ALD;      // 32 x ALD  (Tinv)
  float* betas = Tl   + CHUNKC * ALD;      // 32

  const int tid  = threadIdx.x;
  const int lane = tid & 31;
  const int wave = tid >> 5;
  const int vslice = blockIdx.x;           // 0..3
  const int bh     = blockIdx.y;           // 0..15
  const int vbase  = vslice * DVT;

  const float* qb = qg + (size_t)bh * LSEQ * DK;
  const float* kb = kg + (size_t)bh * LSEQ * DK;
  const float* vb = vg + (size_t)bh * LSEQ * DK;
  const float* bb = bg + (size_t)bh * LSEQ;
  float* ob = outg + (size_t)bh * LSEQ * DK;

  // zero state slice (vectorized)
  for (int i = tid; i < DK * DVT / 4; i += 256) {
    float4 z; z.x = z.y = z.z = z.w = 0.f;
    *(float4*)(S + i * 4) = z;
  }
  __syncthreads();

  const int m16 = lane & 15;          // M (or N) within a 16-wide tile
  const int kh  = (lane >> 4) * 2;    // K sub-offset for A/B operands
  const int mb8 = (lane >> 4) * 8;    // M base for C/D rows

  for (int ci = 0; ci < NCHUNK; ++ci) {
    const int row0 = ci * CHUNKC;
    if (tid < 32) betas[tid] = bb[row0 + tid];

    // ---- P0: l2-normalize q,k rows into LDS (float4 global loads) ----
#pragma unroll 1
    for (int it = 0; it < 4; ++it) {
      int t = wave * 4 + it;
      const float4* qr4 = (const float4*)(qb + (size_t)(row0 + t) * DK);
      const float4* kr4 = (const float4*)(kb + (size_t)(row0 + t) * DK);
      float4 q0 = qr4[lane * 2], q1 = qr4[lane * 2 + 1];
      float4 k0 = kr4[lane * 2], k1 = kr4[lane * 2 + 1];
      float sq = q0.x*q0.x + q0.y*q0.y + q0.z*q0.z + q0.w*q0.w
               + q1.x*q1.x + q1.y*q1.y + q1.z*q1.z + q1.w*q1.w;
      float sk = k0.x*k0.x + k0.y*k0.y + k0.z*k0.z + k0.w*k0.w
               + k1.x*k1.x + k1.y*k1.y + k1.z*k1.z + k1.w*k1.w;
      float rq = rsqrtf(wsum32(sq));
      float rk = rsqrtf(wsum32(sk));
      q0.x*=rq; q0.y*=rq; q0.z*=rq; q0.w*=rq; q1.x*=rq; q1.y*=rq; q1.z*=rq; q1.w*=rq;
      k0.x*=rk; k0.y*=rk; k0.z*=rk; k0.w*=rk; k1.x*=rk; k1.y*=rk; k1.z*=rk; k1.w*=rk;
      float4* qd = (float4*)(qs + t * QLD + lane * 8);
      float4* kd = (float4*)(ks + t * QLD + lane * 8);
      qd[0] = q0; qd[1] = q1; kd[0] = k0; kd[1] = k1;
    }
    __syncthreads();

    // ---- P1: v slice * beta ; A = (k*beta) @ k^T ; prefetch next chunk ----
    for (int idx = tid; idx < CHUNKC * (DVT / 4); idx += 256) {
      int t = idx >> 4, j4 = (idx & 15) * 4;
      float4 vv = *(const float4*)(vb + (size_t)(row0 + t) * DK + vbase + j4);
      float bt = betas[t];
      vv.x *= bt; vv.y *= bt; vv.z *= bt; vv.w *= bt;
      *(float4*)(vs + t * VLD + j4) = vv;
    }

    if (wave < 4) {
      int mt = (wave >> 1) * 16, nt = (wave & 1) * 16;
      v8f acc = {};
      float bm = betas[mt + m16];
      for (int k = 0; k < DK; k += 4) {
        float ax = ks[(mt + m16) * QLD + k + kh] * bm;
        float ay = ks[(mt + m16) * QLD + k + kh + 1] * bm;
        float bx = ks[(nt + m16) * QLD + k + kh];        // B = k^T
        float by = ks[(nt + m16) * QLD + k + kh + 1];
        acc = wmma4(ax, ay, bx, by, acc);
      }
#pragma unroll
      for (int r = 0; r < 8; ++r) Al[(mt + mb8 + r) * ALD + nt + m16] = acc[r];
    }

    if (ci + 1 < NCHUNK) {   // warm L2/WGP$ for the next chunk while we compute
      const float* qn = qb + (size_t)(row0 + CHUNKC) * DK;
      const float* kn = kb + (size_t)(row0 + CHUNKC) * DK;
      __builtin_prefetch(qn + tid * 32, 0, 0);
      __builtin_prefetch(kn + tid * 32, 0, 0);
      if (tid < 64) {
        int r = tid >> 1, h = (tid & 1) * 32;
        __builtin_prefetch(vb + (size_t)(row0 + CHUNKC + r) * DK + vbase + h, 0, 0);
      }
    }
    __syncthreads();

    // ---- P2: blocked Tinv — diagonal 16x16 inverses in parallel ----
    if (wave < 2) {
      int base = wave * 16;
      if (lane < 16) {
        for (int i = 0; i < 16; ++i) {
          float s = (lane == i) ? 1.f : 0.f;
          for (int j = 0; j < i; ++j)
            s -= Al[(base + i) * ALD + base + j] * Tl[(base + j) * ALD + base + lane];
          Tl[(base + i) * ALD + base + lane] = s;
        }
      }
    } else if (wave == 2) {
      for (int idx = lane; idx < 256; idx += 32) {  // zero upper-right block
        int i = idx >> 4, j = idx & 15;
        Tl[i * ALD + 16 + j] = 0.f;
      }
    }
    __syncthreads();

    // ---- P3: X21 = -X22 @ (A21 @ X11) via two 16x16x16 WMMA products (wave 0) ----
    if (wave == 0) {
      v8f acc = {};
      for (int k = 0; k < 16; k += 4) {              // P = A21 @ X11
        float ax = Al[(16 + m16) * ALD + k + kh];
        float ay = Al[(16 + m16) * ALD + k + kh + 1];
        float bx = Tl[(k + kh) * ALD + m16];
        float by = Tl[(k + kh + 1) * ALD + m16];
        acc = wmma4(ax, ay, bx, by, acc);
      }
#pragma unroll
      for (int r = 0; r < 8; ++r) Al[(mb8 + r) * ALD + 16 + m16] = acc[r];  // stash P
      v8f acc2 = {};
      for (int k = 0; k < 16; k += 4) {              // X22 @ P
        float ax = Tl[(16 + m16) * ALD + 16 + k + kh];
        float ay = Tl[(16 + m16) * ALD + 16 + k + kh + 1];
        float bx = Al[(k + kh) * ALD + 16 + m16];
        float by = Al[(k + kh + 1) * ALD + 16 + m16];
        acc2 = wmma4(ax, ay, bx, by, acc2);
      }
#pragma unroll
      for (int r = 0; r < 8; ++r) Tl[(16 + mb8 + r) * ALD + m16] = -acc2[r];
    }
    __syncthreads();

    // ---- P4: u = Tinv@vs (8) ; w = Tinv@(k*beta) (32) ; attn = tril(q@k^T) (4) ----
    for (int job = wave; job < 44; job += 8) {
      if (job < 8) {
        int mt = (job >> 2) * 16, nt = (job & 3) * 16;
        v8f acc = {};
        for (int k = 0; k < CHUNKC; k += 4) {
          float ax = Tl[(mt + m16) * ALD + k + kh];
          float ay = Tl[(mt + m16) * ALD + k + kh + 1];
          float bx = vs[(k + kh) * VLD + nt + m16];
          float by = vs[(k + kh + 1) * VLD + nt + m16];
          acc = wmma4(ax, ay, bx, by, acc);
        }
#pragma unroll
        for (int r = 0; r < 8; ++r) us[(mt + mb8 + r) * VLD + nt + m16] = acc[r];
      } else if (job < 40) {
        int j2 = job - 8;
        int mt = (j2 >> 4) * 16, nt = (j2 & 15) * 16;
        v8f acc = {};
        for (int k = 0; k < CHUNKC; k += 4) {
          float ax = Tl[(mt + m16) * ALD + k + kh];
          float ay = Tl[(mt + m16) * ALD + k + kh + 1];
          float bx = ks[(k + kh) * QLD + nt + m16] * betas[k + kh];
          float by = ks[(k + kh + 1) * QLD + nt + m16] * betas[k + kh + 1];
          acc = wmma4(ax, ay, bx, by, acc);
        }
#pragma unroll
        for (int r = 0; r < 8; ++r) ws[(mt + mb8 + r) * QLD + nt + m16] = acc[r];
      } else {
        int j3 = job - 40;
        int mt = (j3 >> 1) * 16, nt = (j3 & 1) * 16;
        v8f acc = {};
        for (int k = 0; k < DK; k += 4) {
          float ax = qs[(mt + m16) * QLD + k + kh];
          float ay = qs[(mt + m16) * QLD + k + kh + 1];
          float bx = ks[(nt + m16) * QLD + k + kh];      // B = k^T
          float by = ks[(nt + m16) * QLD + k + kh + 1];
          acc = wmma4(ax, ay, bx, by, acc);
        }
#pragma unroll
        for (int r = 0; r < 8; ++r) {
          int gi = mt + mb8 + r, gj = nt + m16;
          Al[gi * ALD + gj] = (gj <= gi) ? acc[r] : 0.f;
        }
      }
    }
    __syncthreads();

    // ---- P5: dual-accumulator K=256 loop sharing B(S) fragments:
    //          accu = w@S (-> u_new), acco = q@S (kept in regs for P6) ----
    const int mt5 = (wave >> 2) * 16, nt5 = (wave & 3) * 16;
    v8f acco = {};
    {
      v8f accu = {};
      for (int k = 0; k < DK; k += 4) {
        float bx = S[(k + kh) * DVT + nt5 + m16];
        float by = S[(k + kh + 1) * DVT + nt5 + m16];
        float wx = ws[(mt5 + m16) * QLD + k + kh];
        float wy = ws[(mt5 + m16) * QLD + k + kh + 1];
        float qx = qs[(mt5 + m16) * QLD + k + kh];
        float qy = qs[(mt5 + m16) * QLD + k + kh + 1];
        accu = wmma4(wx, wy, bx, by, accu);
        acco = wmma4(qx, qy, bx, by, acco);
      }
#pragma unroll
      for (int r = 0; r < 8; ++r) {
        int rr = (mt5 + mb8 + r) * VLD + nt5 + m16;
        un[rr] = us[rr] - accu[r];
      }
    }
    __syncthreads();

    // ---- P6: o = q@S + attn@u_new -> global ; then S += k^T @ u_new ----
    for (int k = 0; k < CHUNKC; k += 4) {
      float ax = Al[(mt5 + m16) * ALD + k + kh];
      float ay = Al[(mt5 + m16) * ALD + k + kh + 1];
      float bx = un[(k + kh) * VLD + nt5 + m16];
      float by = un[(k + kh + 1) * VLD + nt5 + m16];
      acco = wmma4(ax, ay, bx, by, acco);
    }
#pragma unroll
    for (int r = 0; r < 8; ++r)
      ob[(size_t)(row0 + mt5 + mb8 + r) * DK + vbase + nt5 + m16] = acco[r];

    // S update: each wave owns 2 k-tile rows (mt), keeps 4 nt accumulators,
    // reuses the A (k^T) fragment 4x per K-step.
    for (int mj = wave; mj < 16; mj += 8) {
      int mt = mj * 16;
      v8f a0, a1, a2, a3;
#pragma unroll
      for (int r = 0; r < 8; ++r) {
        int rowb = (mt + mb8 + r) * DVT + m16;
        a0[r] = S[rowb];  a1[r] = S[rowb + 16];  a2[r] = S[rowb + 32];  a3[r] = S[rowb + 48];
      }
      for (int k = 0; k < CHUNKC; k += 4) {
        float ax = ks[(k + kh) * QLD + mt + m16];        // A = k^T, reused 4x
        float ay = ks[(k + kh + 1) * QLD + mt + m16];
        int b0 = (k + kh) * VLD + m16, b1 = (k + kh + 1) * VLD + m16;
        a0 = wmma4(ax, ay, un[b0],      un[b1],      a0);
        a1 = wmma4(ax, ay, un[b0 + 16], un[b1 + 16], a1);
        a2 = wmma4(ax, ay, un[b0 + 32], un[b1 + 32], a2);
        a3 = wmma4(ax, ay, un[b0 + 48], un[b1 + 48], a3);
      }
#pragma unroll
      for (int r = 0; r < 8; ++r) {
        int rowb = (mt + mb8 + r) * DVT + m16;
        S[rowb] = a0[r];  S[rowb + 16] = a1[r];  S[rowb + 32] = a2[r];  S[rowb + 48] = a3[r];
      }
    }
    __syncthreads();
  }

  // final state slice -> d_out tail (vectorized)
  float* Sout = Sg + (size_t)bh * DK * DK;
  for (int i = tid; i < DK * (DVT / 4); i += 256) {
    int d = i >> 4, j4 = (i & 15) * 4;
    *(float4*)(Sout + (size_t)d * DK + vbase + j4) = *(float4*)(S + d * DVT + j4);
  }
}

extern "C" void kernel_launch(void* const* d_in, const int* in_sizes, int n_in,
                              void* d_out, int out_size, void* d_ws, size_t ws_size,
                              hipStream_t stream) {
  (void)in_sizes; (void)n_in; (void)out_size; (void)d_ws; (void)ws_size;
  const float* q    = (const float*)d_in[0];
  const float* k    = (const float*)d_in[1];
  const float* v    = (const float*)d_in[2];
  const float* beta = (const float*)d_in[3];
  float* out  = (float*)d_out;
  float* Sout = out + (size_t)16 * LSEQ * DK;   // final state after the attention output

  size_t shmem = (size_t)(DK * DVT + 3 * CHUNKC * QLD + 3 * CHUNKC * VLD +
                          2 * CHUNKC * ALD + 32) * sizeof(float);
  hipFuncSetAttribute((const void*)deltanet_chunk_kernel,
                      hipFuncAttributeMaxDynamicSharedMemorySize, (int)shmem);

  dim3 grid(4, 16);   // (dv slices, b*h)
  deltanet_chunk_kernel<<<grid, 256, shmem, stream>>>(q, k, v, beta, out, Sout);
}